// Sinkhorn_seq_29386166239344
// MI455X (gfx1250) — compile-verified
//
#include <hip/hip_runtime.h>
#include <math.h>

// ---------------------------------------------------------------------------
// Sinkhorn distillation loss for MI455X (gfx1250, wave32).
//   y_s, y_t: [32, 128, 4096] f32.  out: scalar f32.
// Phases: softmax(T=2) -> pairwise L1 cdist -> 10 Sinkhorn iters -> sum(P*W)*1e-3
// ---------------------------------------------------------------------------

typedef __attribute__((ext_vector_type(2))) float v2f;
typedef __attribute__((ext_vector_type(8))) float v8f;

#define B_   32
#define L_   128
#define V_   4096
#define ROW_FLOATS   (L_ * V_)        // 524288 floats per sample
#define W_FLOATS     (L_ * L_)        // 16384 floats per sample

// ---------------------------------------------------------------------------
// Kernel 1: row softmax with temperature 2.  8192 rows (p_s then p_t).
// ---------------------------------------------------------------------------
__global__ __launch_bounds__(256) void softmax_kernel(const float* __restrict__ ys,
                                                      const float* __restrict__ yt,
                                                      float* __restrict__ ps,
                                                      float* __restrict__ pt) {
    __shared__ float red[256];
    const int bid = blockIdx.x;
    const int t   = threadIdx.x;
    const float* src;
    float* dst;
    if (bid < B_ * L_) { src = ys + (size_t)bid * V_;           dst = ps + (size_t)bid * V_; }
    else               { src = yt + (size_t)(bid - B_*L_) * V_; dst = pt + (size_t)(bid - B_*L_) * V_; }

    float v[16];
    float m = -INFINITY;
#pragma unroll
    for (int i = 0; i < 16; ++i) {
        v[i] = src[t + 256 * i];
        m = fmaxf(m, v[i]);
    }
    red[t] = m; __syncthreads();
    for (int s = 128; s > 0; s >>= 1) {
        if (t < s) red[t] = fmaxf(red[t], red[t + s]);
        __syncthreads();
    }
    m = red[0]; __syncthreads();

    float sum = 0.0f;
#pragma unroll
    for (int i = 0; i < 16; ++i) {
        v[i] = expf((v[i] - m) * 0.5f);   // softmax(y/2)
        sum += v[i];
    }
    red[t] = sum; __syncthreads();
    for (int s = 128; s > 0; s >>= 1) {
        if (t < s) red[t] += red[t + s];
        __syncthreads();
    }
    const float inv = 1.0f / red[0];
#pragma unroll
    for (int i = 0; i < 16; ++i) dst[t + 256 * i] = v[i] * inv;
}

// ---------------------------------------------------------------------------
// Kernel 2: W[b,i,j] = sum_v |p_s[b,i,v] - p_t[b,j,v]|   (the heavy phase)
// 256 threads, 64x64 (i,j) tile, 4x4 register tile/thread, Vc=64 chunks.
// Double-buffered LDS staged by GLOBAL_LOAD_ASYNC_TO_LDS_B32 (ASYNCcnt):
// chunk c+1 streams into buffer B while the VALU consumes buffer A.
// LDS tiles transposed [v][row], stride 68 -> compute fetches are b128,
// a-loads broadcast, b-loads sweep all 64 banks exactly once.
// ---------------------------------------------------------------------------
__global__ __launch_bounds__(256) void cdist_kernel(const float* __restrict__ ps,
                                                    const float* __restrict__ pt,
                                                    float* __restrict__ W) {
    __shared__ float lds[2][2][64][68];   // [buffer][S/T][v][row]

    const int t  = threadIdx.x;
    const int b  = blockIdx.z;
    const int it = blockIdx.y;        // i-tile (0..1)
    const int jt = blockIdx.x;        // j-tile (0..1)
    const int tx = t & 15;            // 4 j columns
    const int ty = t >> 4;            // 4 i rows

    const float* Ap = ps + (size_t)b * ROW_FLOATS + (size_t)it * 64 * V_;
    const float* Bp = pt + (size_t)b * ROW_FLOATS + (size_t)jt * 64 * V_;

    // async staging lane map: 32 consecutive v per wave -> one 128B line / instr
    const int v0 = t & 63;
    const int r0 = t >> 6;            // 0..3, rows r0 + 4q
    const unsigned ldsS0     = (unsigned)(uintptr_t)&lds[0][0][v0][r0];
    const unsigned ldsT0     = (unsigned)(uintptr_t)&lds[0][1][v0][r0];
    const unsigned bufStride = (unsigned)sizeof(lds[0]);

    // issue one 64-v chunk (32 async b32 per thread) into buffer bi
    auto issue_chunk = [&](int vb, int bi) {
        unsigned go = (unsigned)(((unsigned)r0 * V_ + (unsigned)vb + (unsigned)v0) * 4u);
        unsigned ls = ldsS0 + (unsigned)bi * bufStride;
        unsigned lt = ldsT0 + (unsigned)bi * bufStride;
#pragma unroll
        for (int q = 0; q < 16; ++q) {
            asm volatile("global_load_async_to_lds_b32 %0, %1, %2"
                         :: "v"(ls), "v"(go), "s"(Ap) : "memory");
            asm volatile("global_load_async_to_lds_b32 %0, %1, %2"
                         :: "v"(lt), "v"(go), "s"(Bp) : "memory");
            go += 4u * V_ * 4u;       // advance 4 rows in global
            ls += 16u;                // advance 4 rows in [v][row] LDS tile
            lt += 16u;
        }
    };

    float acc[4][4];
#pragma unroll
    for (int i = 0; i < 4; ++i)
#pragma unroll
        for (int j = 0; j < 4; ++j) acc[i][j] = 0.0f;

    issue_chunk(0, 0);

    for (int c = 0; c < 64; ++c) {
        const int cur = c & 1;
        if (c + 1 < 64) {
            issue_chunk((c + 1) * 64, 1 - cur);              // prefetch next chunk
            asm volatile("s_wait_asynccnt 32" ::: "memory"); // older 32 done
        } else {
            asm volatile("s_wait_asynccnt 0" ::: "memory");
        }
        __syncthreads();   // buffer `cur` visible to all waves

        const float(*sS)[68] = lds[cur][0];
        const float(*sT)[68] = lds[cur][1];
#pragma unroll 4
        for (int v = 0; v < 64; ++v) {
            const float4 a4 = *(const float4*)&sS[v][ty * 4];   // broadcast
            const float4 b4 = *(const float4*)&sT[v][tx * 4];   // 64-bank sweep
            const float a[4] = {a4.x, a4.y, a4.z, a4.w};
            const float bb[4] = {b4.x, b4.y, b4.z, b4.w};
#pragma unroll
            for (int i = 0; i < 4; ++i)
#pragma unroll
                for (int j = 0; j < 4; ++j)
                    acc[i][j] += fabsf(a[i] - bb[j]);
        }
        __syncthreads();   // done reading `cur` before chunk c+2 overwrites it
    }

    float* Wb = W + (size_t)b * W_FLOATS;
#pragma unroll
    for (int i = 0; i < 4; ++i) {
        float4 o; o.x = acc[i][0]; o.y = acc[i][1]; o.z = acc[i][2]; o.w = acc[i][3];
        *(float4*)(Wb + (size_t)(it * 64 + ty * 4 + i) * L_ + jt * 64 + tx * 4) = o;
    }
}

// ---------------------------------------------------------------------------
// Kernel 3: per-sample Sinkhorn.  One block (8 waves) per sample, P in LDS.
// Row/col sums on the matrix core: V_WMMA_F32_16X16X4_F32 with an all-ones
// B fragment (rowsum = P*1 ; colsum = rowsum(P^T)).
// P padded to stride 132 so WMMA A-fragment gathers are bank-conflict-free.
// ---------------------------------------------------------------------------
#define PST 132

__global__ __launch_bounds__(256) void sinkhorn_kernel(const float* __restrict__ Wg,
                                                       float* __restrict__ losses) {
    __shared__ float P[L_ * PST];
    __shared__ float rowsum[L_];
    __shared__ float colsum[L_];
    __shared__ float red[256];

    const int t    = threadIdx.x;
    const int b    = blockIdx.x;
    const int lane = t & 31;
    const int wv   = t >> 5;                 // 8 waves -> 8 blocks of 16 rows/cols
    const int row  = t >> 1;                 // elementwise mapping: half row/thread
    const int cb   = (t & 1) * 64;
    const float* Wb = Wg + (size_t)b * W_FLOATS;

    // K = max(exp(-W/eps), 1e-30), eps = 0.5  ->  exp(-2W)
#pragma unroll 4
    for (int i = 0; i < 64; i += 4) {
        const float4 w = *(const float4*)(Wb + (size_t)row * L_ + cb + i);
        float* p = &P[row * PST + cb + i];
        p[0] = fmaxf(expf(-2.0f * w.x), 1e-30f);
        p[1] = fmaxf(expf(-2.0f * w.y), 1e-30f);
        p[2] = fmaxf(expf(-2.0f * w.z), 1e-30f);
        p[3] = fmaxf(expf(-2.0f * w.w), 1e-30f);
    }
    __syncthreads();

    // WMMA fragment lane mapping (A 16x4 f32: lanes 0-15 K={0,1}, 16-31 K={2,3})
    const int mrow = wv * 16 + (lane & 15);
    const int koff = (lane >> 4) << 1;
    v2f onesf; onesf[0] = 1.0f; onesf[1] = 1.0f;

    for (int iter = 0; iter < 10; ++iter) {
        // ---- row sums: D = P(16x4 chunks) x ones(4x16), 32 chained WMMAs ----
        v8f acc = {};
        for (int c = 0; c < 32; ++c) {
            const int k = 4 * c + koff;
            const v2f a = *(const v2f*)&P[mrow * PST + k];   // ds_load_b64
            acc = __builtin_amdgcn_wmma_f32_16x16x4_f32(false, a, false, onesf,
                                                        (short)0, acc, false, false);
        }
        if (lane == 0 || lane == 16) {
            const int base = wv * 16 + ((lane >> 4) << 3);
            rowsum[base + 0] = acc[0]; rowsum[base + 1] = acc[1];
            rowsum[base + 2] = acc[2]; rowsum[base + 3] = acc[3];
            rowsum[base + 4] = acc[4]; rowsum[base + 5] = acc[5];
            rowsum[base + 6] = acc[6]; rowsum[base + 7] = acc[7];
        }
        __syncthreads();

        // ---- P /= max(rowsum, 1e-10) ----
        {
            const float rs = 1.0f / fmaxf(rowsum[row], 1e-10f);
            float* Pr = &P[row * PST + cb];
#pragma unroll 4
            for (int i = 0; i < 64; i += 4) {
                float4 p = *(float4*)&Pr[i];
                p.x *= rs; p.y *= rs; p.z *= rs; p.w *= rs;
                *(float4*)&Pr[i] = p;
            }
        }
        __syncthreads();

        // ---- col sums: rowsum of P^T via transposed A-fragment gathers ----
        acc = (v8f){};
        for (int c = 0; c < 32; ++c) {
            const int k = 4 * c + koff;
            v2f a;
            a[0] = P[k * PST + mrow];
            a[1] = P[(k + 1) * PST + mrow];
            acc = __builtin_amdgcn_wmma_f32_16x16x4_f32(false, a, false, onesf,
                                                        (short)0, acc, false, false);
        }
        if (lane == 0 || lane == 16) {
            const int base = wv * 16 + ((lane >> 4) << 3);
            colsum[base + 0] = acc[0]; colsum[base + 1] = acc[1];
            colsum[base + 2] = acc[2]; colsum[base + 3] = acc[3];
            colsum[base + 4] = acc[4]; colsum[base + 5] = acc[5];
            colsum[base + 6] = acc[6]; colsum[base + 7] = acc[7];
        }
        __syncthreads();

        // ---- P /= max(colsum, 1e-10) ----
        {
            float* Pr = &P[row * PST + cb];
#pragma unroll 4
            for (int i = 0; i < 64; i += 4) {
                float4 p = *(float4*)&Pr[i];
                const float4 c4 = *(const float4*)&colsum[cb + i];
                p.x /= fmaxf(c4.x, 1e-10f);
                p.y /= fmaxf(c4.y, 1e-10f);
                p.z /= fmaxf(c4.z, 1e-10f);
                p.w /= fmaxf(c4.w, 1e-10f);
                *(float4*)&Pr[i] = p;
            }
        }
        __syncthreads();
    }

    // ---- loss_b = sum(P * W) ----
    float ls = 0.0f;
#pragma unroll 4
    for (int i = 0; i < 64; i += 4) {
        const float4 w = *(const float4*)(Wb + (size_t)row * L_ + cb + i);
        const float4 p = *(const float4*)&P[row * PST + cb + i];
        ls += p.x * w.x + p.y * w.y + p.z * w.z + p.w * w.w;
    }
    red[t] = ls; __syncthreads();
    for (int s = 128; s > 0; s >>= 1) {
        if (t < s) red[t] += red[t + s];
        __syncthreads();
    }
    if (t == 0) losses[b] = red[0];
}

// ---------------------------------------------------------------------------
// Kernel 4: out = 0.001 * sum_b losses[b]
// ---------------------------------------------------------------------------
__global__ void finalize_kernel(const float* __restrict__ losses, float* __restrict__ out) {
    __shared__ float red[32];
    const int t = threadIdx.x;
    red[t] = losses[t];
    __syncthreads();
    for (int s = 16; s > 0; s >>= 1) {
        if (t < s) red[t] += red[t + s];
        __syncthreads();
    }
    if (t == 0) out[0] = 0.001f * red[0];
}

// ---------------------------------------------------------------------------
extern "C" void kernel_launch(void* const* d_in, const int* in_sizes, int n_in,
                              void* d_out, int out_size, void* d_ws, size_t ws_size,
                              hipStream_t stream) {
    const float* ys = (const float*)d_in[0];
    const float* yt = (const float*)d_in[1];
    float* ws = (float*)d_ws;

    float* ps     = ws;                                    // 32*128*4096
    float* pt     = ws + (size_t)B_ * ROW_FLOATS;          // 32*128*4096
    float* W      = pt + (size_t)B_ * ROW_FLOATS;          // 32*128*128
    float* losses = W  + (size_t)B_ * W_FLOATS;            // 32

    softmax_kernel<<<2 * B_ * L_, 256, 0, stream>>>(ys, yt, ps, pt);
    cdist_kernel<<<dim3(2, 2, B_), 256, 0, stream>>>(ps, pt, W);
    sinkhorn_kernel<<<B_, 256, 0, stream>>>(W, losses);
    finalize_kernel<<<1, 32, 0, stream>>>(losses, (float*)d_out);
}